// _Interaction_51788715655491
// MI455X (gfx1250) — compile-verified
//
#include <hip/hip_runtime.h>
#include <math.h>

typedef __attribute__((ext_vector_type(2))) float v2f;
typedef __attribute__((ext_vector_type(8))) float v8f;
typedef __attribute__((ext_vector_type(4))) unsigned int u32x4;
typedef __attribute__((ext_vector_type(8))) int i32x8;
typedef __attribute__((ext_vector_type(4))) int i32x4;

#define N_NODES 50000
#define N_EDGES 800000
#define HDIM 128
#define GDIM 50
#define GPAD 52
#define FDIM 128
#define NGRAPH 256
#define NCONV 2

__device__ __forceinline__ float ssp(float x) {
  // softplus(x) - ln(2), numerically stable
  float ax = fabsf(x);
  return fmaxf(x, 0.0f) + log1pf(expf(-ax)) - 0.69314718055994530942f;
}

__device__ __forceinline__ v8f wmma4(v2f a, v2f b, v8f c) {
  // D = A(16x4,f32) * B(4x16,f32) + C(16x16,f32)
  return __builtin_amdgcn_wmma_f32_16x16x4_f32(false, a, false, b, (short)0, c,
                                               false, false);
}

// ---------------- Tensor Data Mover: 1-D global -> LDS DMA ----------------
// D# per CDNA5 ISA (8.3/8.4): group0 {count=1, lds_addr,
// global_addr, type=2}; group1 {data_size=3 (8B), dim0=tile0=nelem, dim1=1}.
// nelem = number of 8-byte elements, must be <= 65535.
__device__ __forceinline__ void tdm_load_1d(const void* gptr, unsigned lds_off,
                                            unsigned nelem) {
  unsigned long long ga = (unsigned long long)(uintptr_t)gptr;
  u32x4 g0;
  g0[0] = 1u;                                 // count=1, user descriptor
  g0[1] = lds_off;                            // lds_addr (bytes)
  g0[2] = (unsigned)ga;                       // global_addr[31:0]
  g0[3] = (unsigned)(ga >> 32) | (2u << 30);  // global_addr[56:32] | type=2
  i32x8 g1;
  g1[0] = (int)(3u << 16);                            // data_size = 8 bytes
  g1[1] = (int)((nelem & 0xFFFFu) << 16);             // tensor_dim0[15:0]
  g1[2] = (int)(((nelem >> 16) & 0xFFFFu) | (1u << 16)); // dim0[31:16],dim1=1
  g1[3] = (int)((nelem & 0xFFFFu) << 16);             // tile_dim0 = nelem
  g1[4] = 1;                                          // tile_dim1 = 1
  g1[5] = (int)nelem;                                 // tensor_dim0_stride
  g1[6] = 0;
  g1[7] = 0;
  i32x4 z = {0, 0, 0, 0};
#if __clang_major__ >= 23
  i32x8 z8 = {0, 0, 0, 0, 0, 0, 0, 0};
  __builtin_amdgcn_tensor_load_to_lds(g0, g1, z, z, z8, 0);
#else
  __builtin_amdgcn_tensor_load_to_lds(g0, g1, z, z, 0);
#endif
}

// One wave computes a 16-row x 128-col f32 GEMM tile using pair-packed B:
//   BP[q][n] = (Bt[2q][n], Bt[2q+1][n])  -> lane pair index q = k0/2 + half
template <int K>
__device__ __forceinline__ void gemm16p(const float* __restrict__ X, int row0,
                                        int ldx, const float2* __restrict__ BP,
                                        v8f acc[8]) {
  const int lane = threadIdx.x & 31;
  const int half = lane >> 4;
  const int l15 = lane & 15;
  const float* arow = X + (size_t)(row0 + l15) * ldx + half * 2;
#pragma unroll 4
  for (int k0 = 0; k0 < K; k0 += 4) {
    float2 av = *(const float2*)(arow + k0);
    v2f a;
    a.x = av.x;
    a.y = av.y;
    const float2* bp = BP + (size_t)(k0 / 2 + half) * 128 + l15;
#pragma unroll
    for (int nt = 0; nt < 8; ++nt) {
      float2 bv = bp[nt * 16];
      v2f b;
      b.x = bv.x;
      b.y = bv.y;
      acc[nt] = wmma4(a, b, acc[nt]);
    }
  }
}

// ---------------- weight prep & small elementwise kernels ----------------

// P[q*128+n] = ( A[n][kofs+2q], A[n][kofs+2q+1] ), zero past K. A is [128][K].
__global__ void pack_kernel(const float* __restrict__ A, float2* __restrict__ P,
                            int K, int kofs, int halfKp) {
  int idx = blockIdx.x * blockDim.x + threadIdx.x;
  if (idx >= halfKp * 128) return;
  int q = idx / 128, n = idx % 128;
  int k = kofs + 2 * q;
  float2 v;
  v.x = (k < K) ? A[(size_t)n * K + k] : 0.0f;
  v.y = (k + 1 < K) ? A[(size_t)n * K + k + 1] : 0.0f;
  P[idx] = v;
}

__global__ void cutoff_kernel(const float* __restrict__ ew,
                              float* __restrict__ Cw) {
  int e = blockIdx.x * blockDim.x + threadIdx.x;
  if (e < N_EDGES)
    Cw[e] = 0.5f * (cosf(ew[e] * 0.62831853071795864769f) + 1.0f);
}

__global__ void count_kernel(const int* __restrict__ batch,
                             float* __restrict__ counts) {
  int n = blockIdx.x * blockDim.x + threadIdx.x;
  if (n < N_NODES) atomicAdd(&counts[batch[n]], 1.0f);
}

__global__ void init_gs_kernel(const float* __restrict__ sa,
                               float* __restrict__ gs0,
                               float* __restrict__ gs1) {
  int b = threadIdx.x;
  if (b < NGRAPH) {
    gs0[b] = sa[2 * b];
    gs1[b] = sa[2 * b + 1];
  }
}

__global__ void pool_kernel(const float* __restrict__ h,
                            const int* __restrict__ batch,
                            float* __restrict__ gsum) {
  int n = blockIdx.x * blockDim.x + threadIdx.x;
  if (n >= N_NODES) return;
  const float4* row = (const float4*)(h + (size_t)n * HDIM);
  float s = 0.0f;
#pragma unroll
  for (int i = 0; i < HDIM / 4; ++i) {
    float4 v = row[i];
    s += v.x + v.y + v.z + v.w;
  }
  atomicAdd(&gsum[batch[n]], s);
}

__global__ void state_kernel(const float* __restrict__ gsum,
                             const float* __restrict__ counts,
                             float* __restrict__ gs0, float* __restrict__ gs1) {
  int b = threadIdx.x;
  if (b < NGRAPH) {
    float g = gsum[b] / fmaxf(counts[b], 1.0f);
    gs0[b] = g;
    gs1[b] = g;
  }
}

// ---------------- node-side WMMA GEMM kernels (TDM-staged weights) --------

// h2 = h @ W_h.T + sa0*W[:,0] + sa1*W[:,1] + b   (W staged packed in LDS)
__global__ void __launch_bounds__(256)
lin1_kernel(const float* __restrict__ hin,
            const float2* __restrict__ BP,    // packed [64][128] (cols 2..129)
            const float* __restrict__ W,      // original lin1_w [128][130]
            const float* __restrict__ bias,
            const float* __restrict__ gs0, const float* __restrict__ gs1,
            const int* __restrict__ batch, float* __restrict__ h2) {
  __shared__ float2 s_bp[64 * 128];
  if (threadIdx.x < 32) {
    tdm_load_1d(BP, (unsigned)(uintptr_t)&s_bp[0], 64 * 128);
    __builtin_amdgcn_s_wait_tensorcnt(0);
  }
  __syncthreads();
  int tile = blockIdx.x * (blockDim.x >> 5) + (threadIdx.x >> 5);
  if (tile * 16 >= N_NODES) return;
  int row0 = tile * 16;
  const int lane = threadIdx.x & 31;
  const int half = lane >> 4, l15 = lane & 15;
  v8f acc[8] = {};
  gemm16p<HDIM>(hin, row0, HDIM, s_bp, acc);
  int bidx[8];
#pragma unroll
  for (int r = 0; r < 8; ++r) bidx[r] = batch[row0 + r + half * 8];
#pragma unroll
  for (int nt = 0; nt < 8; ++nt) {
    int n = nt * 16 + l15;
    float w0 = W[(size_t)n * 130], w1 = W[(size_t)n * 130 + 1], bb = bias[n];
#pragma unroll
    for (int r = 0; r < 8; ++r) {
      float y = acc[nt][r] + bb + gs0[bidx[r]] * w0 + gs1[bidx[r]] * w1;
      h2[(size_t)(row0 + r + half * 8) * HDIM + n] = y;
    }
  }
}

// Y = X @ Bt (+ bias) -- conv_lin1 (xf) and the final output layer
__global__ void __launch_bounds__(256)
gemm_kernel(const float* __restrict__ X, const float2* __restrict__ BP,
            const float* __restrict__ bias, float* __restrict__ Y) {
  __shared__ float2 s_bp[64 * 128];
  if (threadIdx.x < 32) {
    tdm_load_1d(BP, (unsigned)(uintptr_t)&s_bp[0], 64 * 128);
    __builtin_amdgcn_s_wait_tensorcnt(0);
  }
  __syncthreads();
  int tile = blockIdx.x * (blockDim.x >> 5) + (threadIdx.x >> 5);
  if (tile * 16 >= N_NODES) return;
  int row0 = tile * 16;
  const int lane = threadIdx.x & 31;
  const int half = lane >> 4, l15 = lane & 15;
  v8f acc[8] = {};
  gemm16p<HDIM>(X, row0, HDIM, s_bp, acc);
#pragma unroll
  for (int nt = 0; nt < 8; ++nt) {
    int n = nt * 16 + l15;
    float bb = bias ? bias[n] : 0.0f;
#pragma unroll
    for (int r = 0; r < 8; ++r)
      Y[(size_t)(row0 + r + half * 8) * HDIM + n] = acc[nt][r] + bb;
  }
}

// hout = h2 + ssp(agg @ Bt + bias)
__global__ void __launch_bounds__(256)
lin2_res_kernel(const float* __restrict__ agg, const float2* __restrict__ BP,
                const float* __restrict__ bias, const float* __restrict__ h2,
                float* __restrict__ hout) {
  __shared__ float2 s_bp[64 * 128];
  if (threadIdx.x < 32) {
    tdm_load_1d(BP, (unsigned)(uintptr_t)&s_bp[0], 64 * 128);
    __builtin_amdgcn_s_wait_tensorcnt(0);
  }
  __syncthreads();
  int tile = blockIdx.x * (blockDim.x >> 5) + (threadIdx.x >> 5);
  if (tile * 16 >= N_NODES) return;
  int row0 = tile * 16;
  const int lane = threadIdx.x & 31;
  const int half = lane >> 4, l15 = lane & 15;
  v8f acc[8] = {};
  gemm16p<FDIM>(agg, row0, FDIM, s_bp, acc);
#pragma unroll
  for (int nt = 0; nt < 8; ++nt) {
    int n = nt * 16 + l15;
    float bb = bias[n];
#pragma unroll
    for (int r = 0; r < 8; ++r) {
      size_t idx = (size_t)(row0 + r + half * 8) * HDIM + n;
      hout[idx] = h2[idx] + ssp(acc[nt][r] + bb);
    }
  }
}

// ---------------- fused edge kernel ----------------
// Per 16-edge tile (one wave):
//   t1  = ssp(edge_attr @ W1 + b1)      WMMA GEMM1 (K=52 padded), B from LDS
//   W   = (t1 @ W2 + b2) * C[e]         WMMA GEMM2 after LDS transpose
//   agg[dst] += W * xf[src]             gather + float atomics
__global__ void __launch_bounds__(128)
edge_kernel(const float* __restrict__ EA,     // [E][50]
            const float* __restrict__ Cw,     // [E]
            const float2* __restrict__ W1P,   // packed [26][128]
            const float* __restrict__ b1,
            const float2* __restrict__ W2P,   // packed [64][128]
            const float* __restrict__ b2,
            const float* __restrict__ xf,     // [N][128]
            const int* __restrict__ esrc, const int* __restrict__ edst,
            float* __restrict__ agg) {
  __shared__ float2 s_w1p[26 * 128];
  __shared__ float2 s_w2p[64 * 128];
  __shared__ float tbuf[4][16][132];  // ld=132 -> conflict-free A reads
  if (threadIdx.x < 32) {
    tdm_load_1d(W1P, (unsigned)(uintptr_t)&s_w1p[0], 26 * 128);
    tdm_load_1d(W2P, (unsigned)(uintptr_t)&s_w2p[0], 64 * 128);
    __builtin_amdgcn_s_wait_tensorcnt(0);
  }
  __syncthreads();

  const int wave = threadIdx.x >> 5;
  const int lane = threadIdx.x & 31;
  const int half = lane >> 4, l15 = lane & 15;
  const int e0 = (blockIdx.x * 4 + wave) * 16;
  const int koff = half * 2;

  // GEMM1: [16 x 50(pad 52)] x [52 x 128]
  v8f acc[8] = {};
  const float* arow = EA + (size_t)(e0 + l15) * GDIM;
  for (int k0 = 0; k0 < GPAD; k0 += 4) {
    int ka = k0 + koff;
    v2f a;
    a.x = (ka < GDIM) ? arow[ka] : 0.0f;
    a.y = (ka + 1 < GDIM) ? arow[ka + 1] : 0.0f;
    const float2* bp = s_w1p + (size_t)(k0 / 2 + half) * 128 + l15;
#pragma unroll
    for (int nt = 0; nt < 8; ++nt) {
      float2 bv = bp[nt * 16];
      v2f b;
      b.x = bv.x;
      b.y = bv.y;
      acc[nt] = wmma4(a, b, acc[nt]);
    }
  }

  // bias + shifted-softplus, D-layout -> LDS (row-major per edge)
  float* tb = &tbuf[wave][0][0];
#pragma unroll
  for (int nt = 0; nt < 8; ++nt) {
    int n = nt * 16 + l15;
    float bb = b1[n];
#pragma unroll
    for (int r = 0; r < 8; ++r)
      tb[(r + half * 8) * 132 + n] = ssp(acc[nt][r] + bb);
  }
  __syncthreads();

  // GEMM2: [16 x 128] x [128 x 128], A from LDS transpose, B from LDS
  v8f acc2[8] = {};
  for (int k0 = 0; k0 < FDIM; k0 += 4) {
    int ka = k0 + koff;
    v2f a;
    a.x = tb[l15 * 132 + ka];
    a.y = tb[l15 * 132 + ka + 1];
    const float2* bp = s_w2p + (size_t)(k0 / 2 + half) * 128 + l15;
#pragma unroll
    for (int nt = 0; nt < 8; ++nt) {
      float2 bv = bp[nt * 16];
      v2f b;
      b.x = bv.x;
      b.y = bv.y;
      acc2[nt] = wmma4(a, b, acc2[nt]);
    }
  }

  // epilogue: cutoff scale, gather xf[src], scatter-add to agg[dst]
  int ss[8], dd[8];
  float cc[8];
#pragma unroll
  for (int r = 0; r < 8; ++r) {
    int e = e0 + r + half * 8;
    ss[r] = esrc[e];
    dd[r] = edst[e];
    cc[r] = Cw[e];
  }
#pragma unroll
  for (int nt = 0; nt < 8; ++nt) {
    int n = nt * 16 + l15;
    float bb = b2[n];
#pragma unroll
    for (int r = 0; r < 8; ++r) {
      float wv = (acc2[nt][r] + bb) * cc[r];
      float val = wv * xf[(size_t)ss[r] * FDIM + n];
      atomicAdd(&agg[(size_t)dd[r] * FDIM + n], val);
    }
  }
}

// ---------------- host orchestration ----------------

extern "C" void kernel_launch(void* const* d_in, const int* in_sizes, int n_in,
                              void* d_out, int out_size, void* d_ws,
                              size_t ws_size, hipStream_t stream) {
  (void)in_sizes; (void)n_in; (void)out_size; (void)ws_size;
  const float* h_in   = (const float*)d_in[0];
  const float* ew     = (const float*)d_in[1];
  const float* ea     = (const float*)d_in[2];
  const float* sa     = (const float*)d_in[3];
  const float* lin1_w = (const float*)d_in[4];
  const float* lin1_b = (const float*)d_in[5];
  const float* mlp_w1 = (const float*)d_in[6];
  const float* mlp_b1 = (const float*)d_in[7];
  const float* mlp_w2 = (const float*)d_in[8];
  const float* mlp_b2 = (const float*)d_in[9];
  const float* cl1_w  = (const float*)d_in[10];
  const float* cl2_w  = (const float*)d_in[11];
  const float* cl2_b  = (const float*)d_in[12];
  const float* out_w  = (const float*)d_in[13];
  const float* out_b  = (const float*)d_in[14];
  const int*   eidx   = (const int*)d_in[15];
  const int*   batch  = (const int*)d_in[16];

  float* ws = (float*)d_ws;
  size_t off = 0;
  float* lin1_p = ws + off; off += (size_t)NCONV * 64 * 128 * 2;
  float* w1p    = ws + off; off += (size_t)NCONV * 26 * 128 * 2;
  float* w2p    = ws + off; off += (size_t)NCONV * 64 * 128 * 2;
  float* cl1p   = ws + off; off += (size_t)NCONV * 64 * 128 * 2;
  float* cl2p   = ws + off; off += (size_t)NCONV * 64 * 128 * 2;
  float* outp   = ws + off; off += (size_t)64 * 128 * 2;
  float* Cw     = ws + off; off += N_EDGES;
  float* counts = ws + off; off += NGRAPH;
  float* gsum   = ws + off; off += NGRAPH;
  float* gs0    = ws + off; off += NGRAPH;
  float* gs1    = ws + off; off += NGRAPH;
  float* hbuf   = ws + off; off += (size_t)N_NODES * HDIM;
  float* h2     = ws + off; off += (size_t)N_NODES * HDIM;
  float* xf     = ws + off; off += (size_t)N_NODES * FDIM;
  float* agg    = ws + off; off += (size_t)N_NODES * FDIM;

  // pack all weights into fragment-pair layout (one-time)
  for (int l = 0; l < NCONV; ++l) {
    pack_kernel<<<(64 * 128 + 255) / 256, 256, 0, stream>>>(
        lin1_w + (size_t)l * 128 * 130, (float2*)(lin1_p + (size_t)l * 64 * 128 * 2),
        130, 2, 64);
    pack_kernel<<<(26 * 128 + 255) / 256, 256, 0, stream>>>(
        mlp_w1 + (size_t)l * 128 * GDIM, (float2*)(w1p + (size_t)l * 26 * 128 * 2),
        GDIM, 0, 26);
    pack_kernel<<<(64 * 128 + 255) / 256, 256, 0, stream>>>(
        mlp_w2 + (size_t)l * 128 * 128, (float2*)(w2p + (size_t)l * 64 * 128 * 2),
        128, 0, 64);
    pack_kernel<<<(64 * 128 + 255) / 256, 256, 0, stream>>>(
        cl1_w + (size_t)l * 128 * 128, (float2*)(cl1p + (size_t)l * 64 * 128 * 2),
        128, 0, 64);
    pack_kernel<<<(64 * 128 + 255) / 256, 256, 0, stream>>>(
        cl2_w + (size_t)l * 128 * 128, (float2*)(cl2p + (size_t)l * 64 * 128 * 2),
        128, 0, 64);
  }
  pack_kernel<<<(64 * 128 + 255) / 256, 256, 0, stream>>>(out_w, (float2*)outp,
                                                          128, 0, 64);

  cutoff_kernel<<<(N_EDGES + 255) / 256, 256, 0, stream>>>(ew, Cw);
  hipMemsetAsync(counts, 0, NGRAPH * sizeof(float), stream);
  count_kernel<<<(N_NODES + 255) / 256, 256, 0, stream>>>(batch, counts);
  init_gs_kernel<<<1, 256, 0, stream>>>(sa, gs0, gs1);

  const int nodeTiles = N_NODES / 16;          // 3125 (exact)
  const int nodeBlocks = (nodeTiles + 7) / 8;  // 8 waves / block
  const int edgeBlocks = N_EDGES / 64;         // 4 waves x 16 edges

  const float* hprev = h_in;
  for (int l = 0; l < NCONV; ++l) {
    lin1_kernel<<<nodeBlocks, 256, 0, stream>>>(
        hprev, (const float2*)(lin1_p + (size_t)l * 64 * 128 * 2),
        lin1_w + (size_t)l * 128 * 130, lin1_b + l * 128, gs0, gs1, batch, h2);
    gemm_kernel<<<nodeBlocks, 256, 0, stream>>>(
        h2, (const float2*)(cl1p + (size_t)l * 64 * 128 * 2), nullptr, xf);
    hipMemsetAsync(agg, 0, (size_t)N_NODES * FDIM * sizeof(float), stream);
    edge_kernel<<<edgeBlocks, 128, 0, stream>>>(
        ea, Cw, (const float2*)(w1p + (size_t)l * 26 * 128 * 2),
        mlp_b1 + l * 128, (const float2*)(w2p + (size_t)l * 64 * 128 * 2),
        mlp_b2 + l * 128, xf, eidx, eidx + N_EDGES, agg);
    lin2_res_kernel<<<nodeBlocks, 256, 0, stream>>>(
        agg, (const float2*)(cl2p + (size_t)l * 64 * 128 * 2), cl2_b + l * 128,
        h2, hbuf);
    hipMemsetAsync(gsum, 0, NGRAPH * sizeof(float), stream);
    pool_kernel<<<(N_NODES + 255) / 256, 256, 0, stream>>>(hbuf, batch, gsum);
    state_kernel<<<1, 256, 0, stream>>>(gsum, counts, gs0, gs1);
    hprev = hbuf;
  }
  gemm_kernel<<<nodeBlocks, 256, 0, stream>>>(
      hbuf, (const float2*)outp, out_b, (float*)d_out);
}